// AlignerNet_77163382440672
// MI455X (gfx1250) — compile-verified
//
#include <hip/hip_runtime.h>
#include <hip/hip_bf16.h>
#include <math.h>

typedef __attribute__((ext_vector_type(2))) float v2f;
typedef __attribute__((ext_vector_type(8))) float v8f;

// D = A(16x4) * B(4x16) + C, fp32 WMMA (CDNA5)
__device__ __forceinline__ v8f wmma_f32(v2f a, v2f b, v8f c) {
    return __builtin_amdgcn_wmma_f32_16x16x4_f32(false, a, false, b, (short)0, c, false, false);
}

#define NEG_BIG (-3.402823466e+38f)

// ---------------------------------------------------------------------------
// Conv1d as GEMM: Y[b,m,t] = act( sum_{c,dt} X[b,c,t+dt-pad] * W[m,c,dt] + bias[m] )
// W row-major (M, K=Cin*KS).  Block = 256 thr = 8 waves along N.
// Each wave computes MT M-tiles (16 rows each) x 16 cols, sharing one B
// fragment across the MT WMMAs -> block tile (16*MT) x 128.
// Grid: (T/128, M/(16*MT), B).
// ---------------------------------------------------------------------------
template<int KS, bool RELU, int MT>
__global__ __launch_bounds__(256)
void conv_gemm(const float* __restrict__ X, const float* __restrict__ W,
               const float* __restrict__ bias, float* __restrict__ Y,
               int Cin, int M, int T)
{
    const int K   = Cin * KS;
    const int pad = KS / 2;
    const int b     = blockIdx.z;
    const int mBase = blockIdx.y * (16 * MT);
    const int tid   = threadIdx.x;
    const int wave  = tid >> 5;
    const int lane  = tid & 31;
    const int l15   = lane & 15;
    const int koff  = (lane < 16) ? 0 : 2;
    const int nBase = blockIdx.x * 128 + wave * 16;

    const float* Xb   = X + (size_t)b * Cin * T;
    const int    aRow = mBase + l15;
    const int    col  = nBase + l15;

    v8f acc[MT];
    #pragma unroll
    for (int mt = 0; mt < MT; ++mt) acc[mt] = (v8f){};

    for (int k = 0; k < K; k += 4) {
        const int ka = k + koff;
        // B fragment (im2col), shared by all MT tiles:
        // v.x row ka, v.y row ka+1, column = nBase + (lane&15)
        v2f bf;
        if (KS == 1) {
            bf.x = Xb[(size_t)ka * T + col];
            bf.y = Xb[(size_t)(ka + 1) * T + col];
            if (k + 8 < K) {  // prefetch 2 k-steps ahead (global_prefetch_b8)
                __builtin_prefetch(&Xb[(size_t)(ka + 8) * T + col], 0, 1);
                __builtin_prefetch(&Xb[(size_t)(ka + 9) * T + col], 0, 1);
            }
        } else {
            int c0 = ka / KS,       d0 = ka - c0 * KS;
            int c1 = (ka + 1) / KS, d1 = (ka + 1) - c1 * KS;
            int t0 = col + d0 - pad;
            int t1 = col + d1 - pad;
            bf.x = (t0 >= 0 && t0 < T) ? Xb[(size_t)c0 * T + t0] : 0.0f;
            bf.y = (t1 >= 0 && t1 < T) ? Xb[(size_t)c1 * T + t1] : 0.0f;
        }
        // A fragments: lanes 0-15 -> K=k,k+1 ; lanes 16-31 -> K=k+2,k+3
        #pragma unroll
        for (int mt = 0; mt < MT; ++mt) {
            v2f a = *reinterpret_cast<const v2f*>(W + (size_t)(aRow + mt * 16) * K + ka);
            acc[mt] = wmma_f32(a, bf, acc[mt]);
        }
    }

    // C/D layout: VGPR r -> row + r + (lane<16 ? 0 : 8), col nBase + (lane&15)
    float* Yb = Y + (size_t)b * M * T;
    #pragma unroll
    for (int mt = 0; mt < MT; ++mt) {
        #pragma unroll
        for (int r = 0; r < 8; ++r) {
            int   row = mBase + mt * 16 + r + ((lane < 16) ? 0 : 8);
            float v   = acc[mt][r] + bias[row];
            if (RELU) v = fmaxf(v, 0.0f);
            Yb[(size_t)row * T + col] = v;
        }
    }
}

// ---------------------------------------------------------------------------
// S[b,t] = sum_c X[b,c,t]^2
// ---------------------------------------------------------------------------
__global__ __launch_bounds__(256)
void sqsum_kernel(const float* __restrict__ X, float* __restrict__ S, int C, int T)
{
    int b = blockIdx.y;
    int t = blockIdx.x * 256 + threadIdx.x;
    const float* Xb = X + (size_t)b * C * T + t;
    float s = 0.0f;
    for (int c = 0; c < C; ++c) { float v = Xb[(size_t)c * T]; s += v * v; }
    S[(size_t)b * T + t] = s;
}

// ---------------------------------------------------------------------------
// Attention: per block: 16 query rows x full Tk=512 keys.
// qk via WMMA (K = C = 80), then d = sqrt(max(q2 + k2 - 2qk, 1e-12)),
// mask, write logp, LDS-staged softmax -> attn.
// Grid: (Tq/16, B), block 256 (8 waves x 64 cols; A fragment reused 4x).
// ---------------------------------------------------------------------------
__global__ __launch_bounds__(256)
void attn_kernel(const float* __restrict__ Q,   // (B, C, Tq)
                 const float* __restrict__ Kt,  // (B, C, Tk)
                 const float* __restrict__ q2s, // (B, Tq)
                 const float* __restrict__ k2s, // (B, Tk)
                 const unsigned char* __restrict__ mask, // (B, Tk) bool
                 float* __restrict__ attn, float* __restrict__ logp,
                 int C, int Tq, int Tk)
{
    __shared__ float tile[16 * 516];  // 516-stride padding vs 64-bank LDS
    const int b      = blockIdx.y;
    const int tqBase = blockIdx.x * 16;
    const int tid    = threadIdx.x;
    const int wave   = tid >> 5;
    const int lane   = tid & 31;
    const int l15    = lane & 15;
    const int koff   = (lane < 16) ? 0 : 2;
    const int nBase0 = wave * 64;   // 8 waves x 64 cols = 512

    const float* Qb = Q  + (size_t)b * C * Tq;
    const float* Kb = Kt + (size_t)b * C * Tk;
    const int aRow  = tqBase + l15;

    v8f acc[4] = {{}, {}, {}, {}};
    for (int k = 0; k < C; k += 4) {
        v2f a;  // A[m][kk] = Q[kk, tqBase+m] (channel-major -> strided rows)
        a.x = Qb[(size_t)(k + koff)     * Tq + aRow];
        a.y = Qb[(size_t)(k + koff + 1) * Tq + aRow];
        if (k + 8 < C) {
            __builtin_prefetch(&Qb[(size_t)(k + koff + 8) * Tq + aRow], 0, 1);
            __builtin_prefetch(&Kb[(size_t)(k + koff + 8) * Tk + nBase0 + l15], 0, 1);
        }
        #pragma unroll
        for (int s = 0; s < 4; ++s) {
            int colv = nBase0 + s * 16 + l15;
            v2f bf;
            bf.x = Kb[(size_t)(k + koff)     * Tk + colv];
            bf.y = Kb[(size_t)(k + koff + 1) * Tk + colv];
            acc[s] = wmma_f32(a, bf, acc[s]);  // A-fragment reused across 4 tiles
        }
    }

    const float* q2b = q2s + (size_t)b * Tq;
    const float* k2b = k2s + (size_t)b * Tk;
    const unsigned char* mb = mask + (size_t)b * Tk;
    float* logpB = logp + ((size_t)b * Tq + tqBase) * Tk;

    #pragma unroll
    for (int s = 0; s < 4; ++s) {
        int   colv = nBase0 + s * 16 + l15;
        float kk   = k2b[colv];
        bool  mOK  = mb[colv] != 0;
        #pragma unroll
        for (int r = 0; r < 8; ++r) {
            int   row = r + ((lane < 16) ? 0 : 8);
            float qq  = q2b[tqBase + row];
            float d2  = fmaxf(qq + kk - 2.0f * acc[s][r], 1e-12f);
            float v   = mOK ? sqrtf(d2) : NEG_BIG;
            tile[row * 516 + colv]         = v;
            logpB[(size_t)row * Tk + colv] = v;
        }
    }
    __syncthreads();

    // softmax: 16 threads per row (row = tid/16), half-wave shfl reductions
    const int row = tid >> 4;
    const int j   = tid & 15;
    const float* trow = &tile[row * 516];

    float m = NEG_BIG;
    for (int c = j; c < Tk; c += 16) m = fmaxf(m, trow[c]);
    #pragma unroll
    for (int off = 8; off; off >>= 1) m = fmaxf(m, __shfl_xor(m, off, 32));

    float ssum = 0.0f;
    for (int c = j; c < Tk; c += 16) ssum += __expf(trow[c] - m);
    #pragma unroll
    for (int off = 8; off; off >>= 1) ssum += __shfl_xor(ssum, off, 32);

    float inv = 1.0f / ssum;
    float* attnB = attn + ((size_t)b * Tq + tqBase + row) * Tk;
    for (int c = j; c < Tk; c += 16) attnB[c] = __expf(trow[c] - m) * inv;
}

// ---------------------------------------------------------------------------
extern "C" void kernel_launch(void* const* d_in, const int* in_sizes, int n_in,
                              void* d_out, int out_size, void* d_ws, size_t ws_size,
                              hipStream_t stream)
{
    (void)in_sizes; (void)n_in; (void)out_size; (void)ws_size;

    const float* queries = (const float*)d_in[0];   // (16,  80, 2048)
    const float* keys    = (const float*)d_in[1];   // (16, 512,  512)
    const unsigned char* mask = (const unsigned char*)d_in[2]; // (16,512) bool
    const float* kw1 = (const float*)d_in[3];       // (1024, 512, 3)
    const float* kb1 = (const float*)d_in[4];
    const float* kw2 = (const float*)d_in[5];       // (80, 1024, 1)
    const float* kb2 = (const float*)d_in[6];
    const float* qw1 = (const float*)d_in[7];       // (160, 80, 3)
    const float* qb1 = (const float*)d_in[8];
    const float* qw2 = (const float*)d_in[9];       // (80, 160, 1)
    const float* qb2 = (const float*)d_in[10];
    const float* qw3 = (const float*)d_in[11];      // (80, 80, 1)
    const float* qb3 = (const float*)d_in[12];

    const int B = 16, Tq = 2048, Tk = 512;

    float* ws  = (float*)d_ws;
    float* k1  = ws;                              // 16*1024* 512
    float* k2  = k1  + (size_t)B * 1024 * Tk;     // 16*  80* 512
    float* q1  = k2  + (size_t)B *   80 * Tk;     // 16* 160*2048
    float* q2  = q1  + (size_t)B *  160 * Tq;     // 16*  80*2048
    float* q3  = q2  + (size_t)B *   80 * Tq;     // 16*  80*2048
    float* q2s = q3  + (size_t)B *   80 * Tq;     // 16*2048
    float* k2s = q2s + (size_t)B * Tq;            // 16* 512

    float* attn = (float*)d_out;
    float* logp = attn + (size_t)B * Tq * Tk;

    // keys path: M=1024 -> MT=4 (64 rows/wave-group); M=80 -> MT=5 (whole M in one wave)
    conv_gemm<3, true , 4><<<dim3(Tk / 128, 1024 / 64, B), 256, 0, stream>>>(keys, kw1, kb1, k1, 512, 1024, Tk);
    conv_gemm<1, false, 5><<<dim3(Tk / 128,          1, B), 256, 0, stream>>>(k1,   kw2, kb2, k2, 1024,  80, Tk);
    // query path: M=160 -> MT=5 (grid.y=2); M=80 -> MT=5
    conv_gemm<3, true , 5><<<dim3(Tq / 128,          2, B), 256, 0, stream>>>(queries, qw1, qb1, q1,  80, 160, Tq);
    conv_gemm<1, true , 5><<<dim3(Tq / 128,          1, B), 256, 0, stream>>>(q1,      qw2, qb2, q2, 160,  80, Tq);
    conv_gemm<1, false, 5><<<dim3(Tq / 128,          1, B), 256, 0, stream>>>(q2,      qw3, qb3, q3,  80,  80, Tq);
    // squared norms
    sqsum_kernel<<<dim3(Tq / 256, B), 256, 0, stream>>>(q3, q2s, 80, Tq);
    sqsum_kernel<<<dim3(Tk / 256, B), 256, 0, stream>>>(k2, k2s, 80, Tk);
    // fused distance-attention + softmax
    attn_kernel<<<dim3(Tq / 16, B), 256, 0, stream>>>(q3, k2, q2s, k2s, mask, attn, logp, 80, Tq, Tk);
}